// IoULoss_pixel_52905407152949
// MI455X (gfx1250) — compile-verified
//
#include <hip/hip_runtime.h>
#include <hip/hip_bf16.h>

// IoU loss: pd/gt (16,256,256) f32, pd_mask/gt_mask (16,32,256,256) i32.
// Memory-bound: 264 MB traffic -> ~11.3 us floor @ 23.3 TB/s.
// WMMA f32 16x16x4 with all-ones B used as a 16-row x 4-pixel reduction engine
// (full fp32 accuracy, D columns replicate the per-row partial sums).

typedef __attribute__((ext_vector_type(2))) float v2f;
typedef __attribute__((ext_vector_type(4))) float v4f;
typedef __attribute__((ext_vector_type(8))) float v8f;
typedef __attribute__((ext_vector_type(4))) int   v4i;

#define HW      65536       // 256*256
#define NCHUNK  64          // chunks per (b, n-group)
#define CHUNK   1024        // hw pixels per chunk (1024*64 = 65536)
#define BN      512         // 16*32 (b,n) pairs

__global__ __launch_bounds__(256)
void iou_partials_kernel(const float* __restrict__ pd,
                         const float* __restrict__ gt,
                         const int*   __restrict__ pdm,
                         const int*   __restrict__ gtm,
                         float*       __restrict__ ws) {
    const int lane = threadIdx.x & 31;
    const int gw   = blockIdx.x * 8 + (threadIdx.x >> 5);   // 2048 waves total
    const int chunk = gw & (NCHUNK - 1);
    const int ng    = (gw >> 6) & 1;        // n-group: rows 0-15 or 16-31
    const int b     = gw >> 7;              // 0..15
    const int row   = lane & 15;            // WMMA M row for this lane
    const int half  = lane >> 4;            // K-slot half: 0 -> K0/K1, 1 -> K2/K3
    const int n     = ng * 16 + row;

    const size_t plane = ((size_t)b * 32 + n) * (size_t)HW;
    const int*   pdm_row = pdm + plane;
    const int*   gtm_row = gtm + plane;
    const float* pd_b    = pd + (size_t)b * HW;
    const float* gt_b    = gt + (size_t)b * HW;
    const int    cbase   = chunk * CHUNK;

    v8f c_and = {};   // sum pd * (pdm & gtm)        -> inter
    v8f c_xor = {};   // sum pd * (pdm & ~gtm)       -> union part 2
    v8f c_gt  = {};   // sum gt * gtm                -> union part 1
    const v2f bones = {1.0f, 1.0f};   // B = ones: D[m][*] += sum_k A[m][k]

    for (int j = 0; j < CHUNK; j += 8) {
        // Super-step covers hw [cbase+j, cbase+j+8). Lane half 0 owns the
        // first int4 (K0,K1 of two WMMAs), half 1 the second (K2,K3).
        const int c0 = cbase + j + half * 4;

        v4i pm = *(const v4i*)(pdm_row + c0);
        v4i gm = *(const v4i*)(gtm_row + c0);
        v4f pv = *(const v4f*)(pd_b   + c0);
        v4f gv = *(const v4f*)(gt_b   + c0);

        // stream-ahead prefetch of the mask planes (speculative, OOB-safe)
        __builtin_prefetch(pdm_row + c0 + 512, 0, 0);
        __builtin_prefetch(gtm_row + c0 + 512, 0, 0);

        float aand[4], axor[4], agt[4];
#pragma unroll
        for (int e = 0; e < 4; ++e) {
            const bool p = pm[e] != 0;
            const bool g = gm[e] != 0;
            aand[e] = (p && g)  ? pv[e] : 0.0f;
            axor[e] = (p && !g) ? pv[e] : 0.0f;
            agt[e]  = g         ? gv[e] : 0.0f;
        }

        v2f a;
        // WMMA #1: elements 0,1 of each half (hw j..j+1 and j+4..j+5)
        a[0] = aand[0]; a[1] = aand[1];
        c_and = __builtin_amdgcn_wmma_f32_16x16x4_f32(false, a, false, bones,
                                                      (short)0, c_and, false, false);
        a[0] = axor[0]; a[1] = axor[1];
        c_xor = __builtin_amdgcn_wmma_f32_16x16x4_f32(false, a, false, bones,
                                                      (short)0, c_xor, false, false);
        a[0] = agt[0];  a[1] = agt[1];
        c_gt  = __builtin_amdgcn_wmma_f32_16x16x4_f32(false, a, false, bones,
                                                      (short)0, c_gt,  false, false);
        // WMMA #2: elements 2,3 of each half (hw j+2..j+3 and j+6..j+7)
        a[0] = aand[2]; a[1] = aand[3];
        c_and = __builtin_amdgcn_wmma_f32_16x16x4_f32(false, a, false, bones,
                                                      (short)0, c_and, false, false);
        a[0] = axor[2]; a[1] = axor[3];
        c_xor = __builtin_amdgcn_wmma_f32_16x16x4_f32(false, a, false, bones,
                                                      (short)0, c_xor, false, false);
        a[0] = agt[2];  a[1] = agt[3];
        c_gt  = __builtin_amdgcn_wmma_f32_16x16x4_f32(false, a, false, bones,
                                                      (short)0, c_gt,  false, false);
    }

    // C/D layout: VGPR m, lanes 0-15 -> M=m (N=lane); lanes 16-31 -> M=m+8.
    // All N columns are identical (B = ones), so lane 0 holds rows 0..7 and
    // lane 16 holds rows 8..15 in c[0..7]. Each wave writes 16 row-partials.
    if ((lane & 15) == 0) {
        const int   rbase = half * 8;
        const size_t base = ((size_t)b * 32 + ng * 16 + rbase) * NCHUNK + chunk;
#pragma unroll
        for (int m = 0; m < 8; ++m) {
            ws[0 * (size_t)BN * NCHUNK + base + (size_t)m * NCHUNK] = c_and[m];
            ws[1 * (size_t)BN * NCHUNK + base + (size_t)m * NCHUNK] = c_xor[m];
            ws[2 * (size_t)BN * NCHUNK + base + (size_t)m * NCHUNK] = c_gt[m];
        }
    }
}

__global__ __launch_bounds__(512)
void iou_finalize_kernel(const float* __restrict__ ws, float* __restrict__ out) {
    __shared__ float red[512];
    const int t = threadIdx.x;            // t == (b*32 + n)
    const float* w0 = ws;
    const float* w1 = ws + (size_t)BN * NCHUNK;
    const float* w2 = ws + 2 * (size_t)BN * NCHUNK;

    float s_and = 0.0f, s_xor = 0.0f, s_gt = 0.0f;
    const int base = t * NCHUNK;
#pragma unroll 8
    for (int c = 0; c < NCHUNK; ++c) {
        s_and += w0[base + c];
        s_xor += w1[base + c];
        s_gt  += w2[base + c];
    }
    const float loss = s_and / (s_gt + s_xor + 1e-6f);

    red[t] = loss;
    __syncthreads();
#pragma unroll
    for (int s = 256; s > 0; s >>= 1) {
        if (t < s) red[t] += red[t + s];
        __syncthreads();
    }
    if (t == 0) out[0] = 1.0f - red[0] / 512.0f;
}

extern "C" void kernel_launch(void* const* d_in, const int* in_sizes, int n_in,
                              void* d_out, int out_size, void* d_ws, size_t ws_size,
                              hipStream_t stream) {
    const float* pd  = (const float*)d_in[0];
    const float* gt  = (const float*)d_in[1];
    const int*   pdm = (const int*)d_in[2];
    const int*   gtm = (const int*)d_in[3];
    float* ws  = (float*)d_ws;   // needs 3*512*64*4 = 384 KB
    float* out = (float*)d_out;

    // 2048 waves = 16 b * 2 n-groups * 64 chunks; 8 waves (256 thr) per block.
    iou_partials_kernel<<<256, 256, 0, stream>>>(pd, gt, pdm, gtm, ws);
    iou_finalize_kernel<<<1, 512, 0, stream>>>(ws, out);
}